// PredictionAggregator_44014824849719
// MI455X (gfx1250) — compile-verified
//
#include <hip/hip_runtime.h>
#include <stdint.h>

#if defined(__HIP_DEVICE_COMPILE__) && !__has_builtin(__builtin_amdgcn_wmma_f32_16x16x4_f32)
#error "missing __builtin_amdgcn_wmma_f32_16x16x4_f32 (device pass)"
#endif

typedef __attribute__((ext_vector_type(2))) float v2f;
typedef __attribute__((ext_vector_type(8))) float v8f;

#define PTS 128   // points per group
#define FDIM 36   // input features
#define FOUT 35   // output features

// order-preserving float <-> uint transform (for unsigned atomicMax on floats)
__device__ __forceinline__ unsigned ordkey(float f) {
  unsigned u = __float_as_uint(f);
  return u ^ ((unsigned)((int)u >> 31) | 0x80000000u);
}
__device__ __forceinline__ float ordinv(unsigned k) {
  unsigned u = (k & 0x80000000u) ? (k ^ 0x80000000u) : ~k;
  return __uint_as_float(u);
}

__global__ void __launch_bounds__(PTS)
pred_agg_kernel(const float* __restrict__ pred, const int* __restrict__ amask,
                float* __restrict__ out) {
  __shared__ __align__(16) float s_feat[PTS * FDIM];   // 18432 B, reused as result stage
  __shared__ __align__(16) int s_key[PTS];
  __shared__ __align__(16) int s_rep[PTS];
  __shared__ __align__(16) int s_cid[PTS];
  __shared__ __align__(16) float s_btab[5 * PTS];      // rows: 1, wx, wy, wz, 0
  __shared__ __align__(16) float s_sums[4 * PTS];      // rows: cnt, sx, sy, sz
  __shared__ __align__(16) unsigned s_max[PTS * 33];   // [cluster][conf-logit, 32 logits]

  const int t = threadIdx.x;
  const int bg = blockIdx.x;
  const float* gsrc = pred + (size_t)bg * (PTS * FDIM);

  // ---- Phase 0: async global->LDS staging of the 128x36 tile (9 x B128 per thread)
  {
    unsigned lds0 = (unsigned)(uintptr_t)(&s_feat[0]);  // low 32 bits = LDS byte offset
#pragma unroll
    for (int i = 0; i < 9; ++i) {
      const int idx = i * PTS + t;                       // 16-byte chunk index (0..1151)
      unsigned loff = lds0 + idx * 16;
      uint64_t gaddr = (uint64_t)(uintptr_t)(gsrc + idx * 4);
      asm volatile("global_load_async_to_lds_b128 %0, %1, off"
                   :: "v"(loff), "v"(gaddr) : "memory");
    }
  }
  // init max buffers + constant btab rows while copies are in flight
  for (int i = t; i < PTS * 33; i += PTS) s_max[i] = 0u;
  s_btab[0 * PTS + t] = 1.0f;   // count column
  s_btab[4 * PTS + t] = 0.0f;   // zero row for unused WMMA columns 4..15
  const int active = amask[(size_t)bg * PTS + t] != 0;
  asm volatile("s_wait_asynccnt 0" ::: "memory");
  __syncthreads();

  // ---- Phase 1: voxel key + confidence-weighted coords
  const float x = s_feat[t * FDIM + 0];
  const float y = s_feat[t * FDIM + 1];
  const float z = s_feat[t * FDIM + 2];
  int key;
  if (active) {
    int vx = (int)floorf(x / 0.05f); vx = vx < -128 ? -128 : (vx > 127 ? 127 : vx); vx += 128;
    int vy = (int)floorf(y / 0.05f); vy = vy < -128 ? -128 : (vy > 127 ? 127 : vy); vy += 128;
    int vz = (int)floorf(z / 0.05f); vz = vz < -128 ? -128 : (vz > 127 ? 127 : vz); vz += 128;
    key = (vx * 256 + vy) * 256 + vz;
  } else {
    key = 0x7FFFFFFF;  // sentinel, never forms a cluster
  }
  s_key[t] = key;
  const float conf = 1.0f / (1.0f + __expf(-s_feat[t * FDIM + 3]));
  s_btab[1 * PTS + t] = x * conf;
  s_btab[2 * PTS + t] = y * conf;
  s_btab[3 * PTS + t] = z * conf;
  __syncthreads();

  // ---- Phase 2: representative = first occurrence of key among active points
  int rep = active;
  for (int j = 0; j < t; ++j)
    if (s_key[j] == key) rep = 0;
  s_rep[t] = rep;
  __syncthreads();

  // ---- Phase 3: cluster id = rank of key among distinct keys (sorted order)
  int pos = 0;
  for (int j = 0; j < PTS; ++j)
    pos += (s_rep[j] && (s_key[j] < key)) ? 1 : 0;
  s_cid[t] = active ? pos : 255;  // 255 => member of no cluster
  __syncthreads();

  // ---- Phase 4a: per-cluster maxima via LDS atomics (order-preserving uint max)
  if (active) {
    atomicMax(&s_max[pos * 33 + 0], ordkey(s_feat[t * FDIM + 3]));
#pragma unroll 4
    for (int f = 0; f < 32; ++f)
      atomicMax(&s_max[pos * 33 + 1 + f], ordkey(s_feat[t * FDIM + 4 + f]));
  }

  // ---- Phase 4b: segment sums via one-hot WMMA matmul (branch-free fragments)
  // D[16x16] += A[16x4] (one-hot membership) x B[4x16] (cols: 1, wx, wy, wz, 0...)
  {
    const int lane = t & 31;
    const int wv = t >> 5;
    const int col = lane & 15;
    const int khalf = (lane < 16) ? 0 : 2;        // A/B K-split across lane halves
    const int mb0 = wv * 32;                      // this wave's cluster chunks
    const int m0 = mb0 + col;
    const int m1 = mb0 + 16 + col;
    const float* bp = s_btab + (col < 4 ? col : 4) * PTS;  // lane's B column source row
    v8f acc0 = {};
    v8f acc1 = {};
#pragma unroll 4
    for (int kk = 0; kk < 32; ++kk) {
      const int p0 = kk * 4 + khalf;              // even -> aligned b64 pair loads
      const v2f b = *(const v2f*)(bp + p0);
      const int2 cp = *(const int2*)(s_cid + p0);
      v2f a0, a1;
      a0.x = (cp.x == m0) ? 1.0f : 0.0f;
      a0.y = (cp.y == m0) ? 1.0f : 0.0f;
      a1.x = (cp.x == m1) ? 1.0f : 0.0f;
      a1.y = (cp.y == m1) ? 1.0f : 0.0f;
#if defined(__HIP_DEVICE_COMPILE__)
      acc0 = __builtin_amdgcn_wmma_f32_16x16x4_f32(false, a0, false, b, (short)0, acc0,
                                                   false, false);
      acc1 = __builtin_amdgcn_wmma_f32_16x16x4_f32(false, a1, false, b, (short)0, acc1,
                                                   false, false);
#endif
    }
    // D layout: VGPR v, lane L -> row = v + 8*(L>=16), col = L&15; cols 0..3 carry data
    const int rowoff = (lane < 16) ? 0 : 8;
    if (col < 4) {
      float* sp = s_sums + col * PTS;
#pragma unroll
      for (int v = 0; v < 8; ++v) {
        sp[mb0 + v + rowoff] = acc0[v];
        sp[mb0 + 16 + v + rowoff] = acc1[v];
      }
    }
  }
  __syncthreads();

  // ---- Phase 5: assemble output row t (cluster t of this group; zeros if absent)
  float* s_res = s_feat;  // reuse tile LDS as 128x35 staging buffer
  {
    const int c = t;
    const float cnt = s_sums[0 * PTS + c];
    const bool has = cnt > 0.5f;
    const float inv = has ? (1.0f / cnt) : 0.0f;  // ref: / max(count, 1)
    s_res[c * FOUT + 0] = s_sums[1 * PTS + c] * inv;
    s_res[c * FOUT + 1] = s_sums[2 * PTS + c] * inv;
    s_res[c * FOUT + 2] = s_sums[3 * PTS + c] * inv;
    float mc = 0.0f;
    if (has) {
      const float m3 = ordinv(s_max[c * 33 + 0]);   // max raw conf logit
      mc = 1.0f / (1.0f + __expf(-m3));             // sigmoid is monotone => max conf
    }
    s_res[c * FOUT + 3] = mc;
    for (int f = 0; f < 32; ++f)
      s_res[c * FOUT + 4 + f] = has ? ordinv(s_max[c * 33 + 1 + f]) : 0.0f;
  }
  __syncthreads();

  // ---- Phase 6: coalesced streaming store of the group's 128x35 slab
  float* gdst = out + (size_t)bg * (PTS * FOUT);
  for (int i = t; i < PTS * FOUT; i += PTS) gdst[i] = s_res[i];
}

extern "C" void kernel_launch(void* const* d_in, const int* in_sizes, int n_in,
                              void* d_out, int out_size, void* d_ws, size_t ws_size,
                              hipStream_t stream) {
  (void)n_in; (void)out_size; (void)d_ws; (void)ws_size;
  const float* pred = (const float*)d_in[0];
  const int* amask = (const int*)d_in[1];
  float* out = (float*)d_out;
  const int groups = in_sizes[1] / PTS;  // B*G = 8192
  pred_agg_kernel<<<dim3(groups), dim3(PTS), 0, stream>>>(pred, amask, out);
}